// WindowAttention_60129542718
// MI455X (gfx1250) — compile-verified
//
#include <hip/hip_runtime.h>
#include <hip/hip_bf16.h>

typedef __attribute__((ext_vector_type(2))) float v2f;
typedef __attribute__((ext_vector_type(8))) float v8f;

#define NUM_HEADS 6
#define HEAD_DIM 32
#define NTOK 64
#define META_HIDDEN 256
#define QKV_STRIDE 576   // 3*192 floats per token
#define ROW_STRIDE 36    // 32 + 4 pad floats (conflict-free fragment reads)
#define P_STRIDE 68      // 64 + 4 pad floats

// SCALE = 32^-0.5; KS = SCALE*log2(e); bias pre-multiplied by 1/SCALE in ws.
#define KS        0.25500526954754613f
#define INV_SCALE 5.656854249492381f

static __device__ __forceinline__ v8f wmma_f32_4(v2f a, v2f b, v8f c) {
    return __builtin_amdgcn_wmma_f32_16x16x4_f32(false, a, false, b, (short)0, c,
                                                 false, false);
}

// Butterfly sum over the 16-lane half-wave using ds_swizzle_b32
// (group-of-32 xor mode: offset = xor_mask<<10 | and_mask 0x1F).
static __device__ __forceinline__ float red_sum16(float x) {
    x += __int_as_float(__builtin_amdgcn_ds_swizzle(__float_as_int(x), 0x041F));
    x += __int_as_float(__builtin_amdgcn_ds_swizzle(__float_as_int(x), 0x081F));
    x += __int_as_float(__builtin_amdgcn_ds_swizzle(__float_as_int(x), 0x101F));
    x += __int_as_float(__builtin_amdgcn_ds_swizzle(__float_as_int(x), 0x201F));
    return x;
}

// ---------------- bias MLP: [64,64,2] -> Linear(2,256) -> ReLU -> Linear(256,6) ----------------
// Output written pre-scaled by 1/SCALE, in WMMA C-fragment layout:
//   ws[h][wave][j][hi][lo][v]  (v contiguous, 8 floats = 32B per lane)
// so the attention kernel loads each 16x16 C tile as two b128 loads per lane.
__global__ __launch_bounds__(256) void bias_mlp_kernel(
        const float* __restrict__ rel_pos,  // [64,64,2]
        const float* __restrict__ w1,       // [2,256]
        const float* __restrict__ b1,       // [256]
        const float* __restrict__ w2,       // [256,6]
        const float* __restrict__ b2,       // [6]
        float* __restrict__ bias)           // [6,4,4,2,16,8] fragment layout
{
    int idx = blockIdx.x * 256 + threadIdx.x;   // 0..4095 = m*64+n
    float x0 = rel_pos[idx * 2 + 0];
    float x1 = rel_pos[idx * 2 + 1];
    float acc[NUM_HEADS];
    #pragma unroll
    for (int h = 0; h < NUM_HEADS; ++h) acc[h] = b2[h];
    for (int k = 0; k < META_HIDDEN; ++k) {
        float hv = fmaf(x0, w1[k], fmaf(x1, w1[META_HIDDEN + k], b1[k]));
        hv = fmaxf(hv, 0.0f);
        #pragma unroll
        for (int h = 0; h < NUM_HEADS; ++h)
            acc[h] = fmaf(hv, w2[k * NUM_HEADS + h], acc[h]);
    }
    const int m = idx >> 6, n = idx & 63;
    const int v = m & 7, hi = (m >> 3) & 1, wav = m >> 4;
    const int j = n >> 4, lo = n & 15;
    #pragma unroll
    for (int h = 0; h < NUM_HEADS; ++h) {
        int fidx = ((((h * 4 + wav) * 4 + j) * 2 + hi) * 16 + lo) * 8 + v;
        bias[fidx] = acc[h] * INV_SCALE;
    }
}

// ---------------- fused window attention: one block = one (window, head) ----------------
__global__ __launch_bounds__(128) void window_attn_kernel(
        const float* __restrict__ qkv,   // [2048, 64, 576]
        const float* __restrict__ bias,  // fragment layout, pre-scaled by 1/SCALE
        float* __restrict__ out)         // [2048, 64, 192]
{
    __shared__ float q_lds[NTOK][ROW_STRIDE];
    __shared__ float k_lds[NTOK][ROW_STRIDE];
    __shared__ float v_lds[NTOK][ROW_STRIDE];
    __shared__ float p_lds[NTOK][P_STRIDE];

    const int bh = blockIdx.x;
    const int b  = bh / NUM_HEADS;
    const int h  = bh % NUM_HEADS;
    const int tid = threadIdx.x;

    // ---- stage Q,K,V tiles (64x32 each) into LDS, float4 loads ----
    const float* base = qkv + (size_t)b * (NTOK * QKV_STRIDE) + h * HEAD_DIM;
    #pragma unroll
    for (int t = 0; t < 4; ++t) {
        int idx = tid + t * 128;       // 0..511
        int row = idx >> 3;            // token
        int c4  = (idx & 7) * 4;       // float offset within head_dim
        const float* src = base + row * QKV_STRIDE + c4;
        float4 qv = *(const float4*)(src);
        float4 kv = *(const float4*)(src + 192);
        float4 vv = *(const float4*)(src + 384);
        *(float4*)&q_lds[row][c4] = qv;
        *(float4*)&k_lds[row][c4] = kv;
        *(float4*)&v_lds[row][c4] = vv;
    }

    const int lane = tid & 31;
    const int wave = tid >> 5;          // 4 waves, wave w owns rows [16w,16w+16)
    const int lo = lane & 15;
    const int hi = lane >> 4;           // half-wave select

    // ---- init accumulators with bias/SCALE as the WMMA C operand (b128 x2 per tile) ----
    const float* bfrag = bias + ((((h * 4 + wave) * 4) * 2 + hi) * 16 + lo) * 8;
    v8f acc[4];
    #pragma unroll
    for (int j = 0; j < 4; ++j)
        acc[j] = *(const v8f*)(bfrag + j * 256);

    __syncthreads();

    // ---- QK^T (+bias via C): 4 col tiles of 16x16, K=32 in 8 wmma_f32_16x16x4 steps ----
    const int rowA = wave * 16 + lo;
    #pragma unroll
    for (int kk = 0; kk < 8; ++kk) {
        const int kidx = kk * 4 + hi * 2;
        v2f a;
        a.x = q_lds[rowA][kidx];
        a.y = q_lds[rowA][kidx + 1];
        #pragma unroll
        for (int j = 0; j < 4; ++j) {
            v2f bf;
            bf.x = k_lds[j * 16 + lo][kidx];
            bf.y = k_lds[j * 16 + lo][kidx + 1];
            acc[j] = wmma_f32_4(a, bf, acc[j]);
        }
    }

    // ---- softmax: x = exp2(KS * z); logits bounded (|KS*z| ~ <=10) -> no max pass;
    //      normalization deferred to the output (P@V is linear in rows) ----
    float rr[8];
    #pragma unroll
    for (int v = 0; v < 8; ++v) {
        const int m = wave * 16 + v + hi * 8;   // row of attn matrix
        float x0 = __builtin_amdgcn_exp2f(acc[0][v] * KS);
        float x1 = __builtin_amdgcn_exp2f(acc[1][v] * KS);
        float x2 = __builtin_amdgcn_exp2f(acc[2][v] * KS);
        float x3 = __builtin_amdgcn_exp2f(acc[3][v] * KS);
        float s = red_sum16((x0 + x1) + (x2 + x3));
        rr[v] = __builtin_amdgcn_rcpf(s);       // v_rcp_f32: ample for softmax norm
        // stage un-normalized probs to wave-private LDS strip (C-layout -> memory)
        p_lds[m][ 0 + lo] = x0;
        p_lds[m][16 + lo] = x1;
        p_lds[m][32 + lo] = x2;
        p_lds[m][48 + lo] = x3;
    }
    // DS ops from the same wave are in-order; strip is wave-private -> no barrier.

    // ---- P @ V: 2 output tiles of 16x16, K=64 in 16 steps ----
    v8f o0 = (v8f)(0.0f);
    v8f o1 = (v8f)(0.0f);
    #pragma unroll
    for (int kk = 0; kk < 16; ++kk) {
        const int kidx = kk * 4 + hi * 2;
        v2f a;
        a.x = p_lds[rowA][kidx];
        a.y = p_lds[rowA][kidx + 1];
        v2f bf0, bf1;
        bf0.x = v_lds[kidx][ 0 + lo];
        bf0.y = v_lds[kidx + 1][ 0 + lo];
        bf1.x = v_lds[kidx][16 + lo];
        bf1.y = v_lds[kidx + 1][16 + lo];
        o0 = wmma_f32_4(a, bf0, o0);
        o1 = wmma_f32_4(a, bf1, o1);
    }

    // ---- store with deferred softmax normalization: out[b][m][h*32 + d] ----
    float* obase = out + (size_t)b * (NTOK * 192) + h * HEAD_DIM;
    #pragma unroll
    for (int v = 0; v < 8; ++v) {
        const int m = wave * 16 + v + hi * 8;
        obase[m * 192 + lo]      = o0[v] * rr[v];
        obase[m * 192 + 16 + lo] = o1[v] * rr[v];
    }
}

extern "C" void kernel_launch(void* const* d_in, const int* in_sizes, int n_in,
                              void* d_out, int out_size, void* d_ws, size_t ws_size,
                              hipStream_t stream) {
    const float* qkv     = (const float*)d_in[0];
    const float* rel_pos = (const float*)d_in[1];
    const float* w1      = (const float*)d_in[2];
    const float* b1      = (const float*)d_in[3];
    const float* w2      = (const float*)d_in[4];
    const float* b2      = (const float*)d_in[5];
    float* out  = (float*)d_out;
    float* bias = (float*)d_ws;   // 6*4*4*2*16*8 floats = 96 KB scratch

    bias_mlp_kernel<<<16, 256, 0, stream>>>(rel_pos, w1, b1, w2, b2, bias);
    window_attn_kernel<<<2048 * NUM_HEADS, 128, 0, stream>>>(qkv, bias, out);
}